// HungarianLossCustom_84189948936345
// MI455X (gfx1250) — compile-verified
//
#include <hip/hip_runtime.h>
#include <hip/hip_bf16.h>
#include <math.h>

// Problem constants (from reference)
#define BB 128
#define NN 256
#define NUM_CLASSES 8
#define CE_COEFF 10.0f

typedef __attribute__((ext_vector_type(2))) float v2f;
typedef __attribute__((ext_vector_type(8))) float v8f;

// ---------------- Shared-memory layout (dynamic LDS) ----------------
// cost : float [256*256]          262144 B   (whole dist matrix in LDS)
// u,v,minv : double[257] each       3*2056 B (JV dual variables, f64 like ref)
// p,way,used : int[257] each        3*1028 B
// col  : int[256]                    1024 B
// redM,redN : float[256] each        2048 B
// redC : int[256]                    1024 B
#define SM_COST_BYTES   (NN * NN * 4)
#define SM_TOTAL_BYTES  (SM_COST_BYTES + 3*257*8 + 3*257*4 + 256*4 + 2*256*4 + 256*4)

__global__ __launch_bounds__(256, 1)
void hungarian_loss_kernel(const float* __restrict__ set1,
                           const float* __restrict__ set2,
                           float* __restrict__ partial) {
    extern __shared__ unsigned char smem_raw[];
    float*  cost_sh = (float*)smem_raw;
    double* u_sh    = (double*)(smem_raw + SM_COST_BYTES);
    double* v_sh    = u_sh + 257;
    double* minv_sh = v_sh + 257;
    int*    p_sh    = (int*)(minv_sh + 257);
    int*    way_sh  = p_sh + 257;
    int*    used_sh = way_sh + 257;
    int*    col_sh  = used_sh + 257;
    float*  redM    = (float*)(col_sh + 256);
    float*  redN    = redM + 256;
    int*    redC    = (int*)(redN + 256);

    const int b    = blockIdx.x;
    const int tid  = threadIdx.x;
    const int lane = tid & 31;
    const int wave = tid >> 5;

    const float* s1 = set1 + (size_t)b * NN * 3;
    const float* s2 = set2 + (size_t)b * NN * (2 + NUM_CLASSES);

    // ---------------- Phase 1: dist matrix via WMMA (rank-2 GEMM) ----------
    // dist^2[m][n] = |p1[m]|^2 + |p2[n]|^2 - 2*dot(p1[m],p2[n])
    // A (16x4 f32): lane<16 holds K=0,1 in a[0],a[1] (pre-scaled by -2);
    // lanes 16-31 hold K=2,3 -> zero padded.  B (4x16) symmetric layout.
    for (int t = wave; t < 256; t += 8) {
        const int mt = (t >> 4) << 4;   // tile row base   (set1 index)
        const int nt = (t & 15) << 4;   // tile col base   (set2 index)

        v2f a; a[0] = 0.0f; a[1] = 0.0f;
        v2f bm; bm[0] = 0.0f; bm[1] = 0.0f;
        if (lane < 16) {
            const float* pa = s1 + (size_t)(mt + lane) * 3;
            a[0] = -2.0f * pa[0];
            a[1] = -2.0f * pa[1];
            const float* pb = s2 + (size_t)(nt + lane) * (2 + NUM_CLASSES);
            bm[0] = pb[0];
            bm[1] = pb[1];
        }
        v8f c = {};
        // D = A * B + C   (16x16 f32 accum, 8 VGPRs)
        c = __builtin_amdgcn_wmma_f32_16x16x4_f32(
                /*neg_a=*/false, a, /*neg_b=*/false, bm,
                /*c_mod=*/(short)0, c, /*reuse_a=*/false, /*reuse_b=*/false);

        const int nn   = nt + (lane & 15);
        const int moff = (lane >= 16) ? 8 : 0;
        const float* qb = s2 + (size_t)nn * (2 + NUM_CLASSES);
        const float nrm2 = qb[0] * qb[0] + qb[1] * qb[1];
        #pragma unroll
        for (int r = 0; r < 8; ++r) {
            const int mm = mt + moff + r;
            const float* qa = s1 + (size_t)mm * 3;
            const float nrm1 = qa[0] * qa[0] + qa[1] * qa[1];
            const float ss = c[r] + nrm1 + nrm2;
            cost_sh[mm * NN + nn] = sqrtf(fmaxf(ss, 0.0f));
        }
    }
    __syncthreads();

    // ---------------- Phase 2: Hungarian (JV) on wave 0 only ---------------
    // Single wave32 in lockstep -> no block barriers needed; LDS ops from a
    // wave are in-order so producer/consumer within the wave is safe.
    if (tid < 32) {
        const int l = lane;
        for (int j = l; j < 257; j += 32) {
            u_sh[j] = 0.0;  v_sh[j] = 0.0;  p_sh[j] = 0;
        }
        for (int i = 1; i <= NN; ++i) {
            if (l == 0) p_sh[0] = i;
            for (int j = l; j < 257; j += 32) { minv_sh[j] = 1e18; used_sh[j] = 0; }
            int j0 = 0;
            while (true) {
                if (l == 0) used_sh[j0] = 1;
                const int    i0  = p_sh[j0];
                const double ui0 = u_sh[i0];

                double bestv = 1e30;
                int    bestj = 0x40000000;
                #pragma unroll
                for (int k = 0; k < 8; ++k) {
                    const int j = 1 + l + 32 * k;
                    if (!used_sh[j]) {
                        const double cur =
                            (double)cost_sh[(i0 - 1) * NN + (j - 1)] - ui0 - v_sh[j];
                        if (cur < minv_sh[j]) { minv_sh[j] = cur; way_sh[j] = j0; }
                        const double mv = minv_sh[j];
                        if (mv < bestv) { bestv = mv; bestj = j; } // strict '<' keeps lowest j
                    }
                }
                // cross-lane argmin, lowest-index tie-break (matches np.argmin)
                #pragma unroll
                for (int off = 16; off; off >>= 1) {
                    const double ov = __shfl_xor(bestv, off, 32);
                    const int    oj = __shfl_xor(bestj, off, 32);
                    if (ov < bestv || (ov == bestv && oj < bestj)) { bestv = ov; bestj = oj; }
                }
                const int    j1    = bestj;
                const double delta = bestv;
                #pragma unroll
                for (int k = 0; k < 8; ++k) {
                    const int j = 1 + l + 32 * k;
                    if (used_sh[j]) { u_sh[p_sh[j]] += delta; v_sh[j] -= delta; }
                    else            { minv_sh[j] -= delta; }
                }
                if (l == 0) { u_sh[p_sh[0]] += delta; v_sh[0] -= delta; }
                j0 = j1;
                if (p_sh[j0] == 0) break;
            }
            if (l == 0) {          // augmenting path rewrite (serial)
                int jj = j0;
                while (jj) { const int jp = way_sh[jj]; p_sh[jj] = p_sh[jp]; jj = jp; }
            }
        }
        for (int j = 1 + l; j <= NN; j += 32) col_sh[p_sh[j] - 1] = j - 1;
    }
    __syncthreads();

    // ---------------- Phase 3: per-row loss terms --------------------------
    {
        const int i = tid;                       // row in set1
        const int c = col_sh[i];                 // matched col in set2
        const float matched = cost_sh[i * NN + c];

        const int tgt = (int)s1[(size_t)i * 3 + 2];
        const float* lg = s2 + (size_t)c * (2 + NUM_CLASSES) + 2;
        float mx = lg[0];
        #pragma unroll
        for (int r = 1; r < NUM_CLASSES; ++r) mx = fmaxf(mx, lg[r]);
        float sum = 0.0f;
        #pragma unroll
        for (int r = 0; r < NUM_CLASSES; ++r) sum += expf(lg[r] - mx);
        const float lse = mx + logf(sum);
        const int   msk = (tgt != -1) ? 1 : 0;
        const int   st  = msk ? tgt : 0;
        const float nll = -(lg[st] - lse);

        redM[i] = matched;
        redN[i] = msk ? nll : 0.0f;
        redC[i] = msk;
    }
    __syncthreads();

    if (tid == 0) {   // fixed-order serial reduce -> deterministic
        float sm = 0.0f, sn = 0.0f; int cnt = 0;
        for (int i = 0; i < NN; ++i) { sm += redM[i]; sn += redN[i]; cnt += redC[i]; }
        const float denom = (float)(cnt > 1 ? cnt : 1);
        partial[b] = sm / (float)NN + CE_COEFF * (sn / denom);
    }
}

// Deterministic final reduce over the 128 per-batch partials.
__global__ void final_sum_kernel(const float* __restrict__ partial,
                                 float* __restrict__ out) {
    if (threadIdx.x == 0 && blockIdx.x == 0) {
        float acc = 0.0f;
        for (int b = 0; b < BB; ++b) acc += partial[b];
        out[0] = acc;
    }
}

extern "C" void kernel_launch(void* const* d_in, const int* in_sizes, int n_in,
                              void* d_out, int out_size, void* d_ws, size_t ws_size,
                              hipStream_t stream) {
    const float* set1 = (const float*)d_in[0];   // [128,256,3]
    const float* set2 = (const float*)d_in[1];   // [128,256,10]
    float* out     = (float*)d_out;              // scalar
    float* partial = (float*)d_ws;               // 128 floats of scratch

    hungarian_loss_kernel<<<dim3(BB), dim3(256), SM_TOTAL_BYTES, stream>>>(
        set1, set2, partial);
    final_sum_kernel<<<dim3(1), dim3(32), 0, stream>>>(partial, out);
}